// GroupedKANLayer_26912265076911
// MI455X (gfx1250) — compile-verified
//
#include <hip/hip_runtime.h>

// ---- problem constants (from reference) ----
#define NCH        31            // NUM_CHANNELS
#define INDIM      7             // 2*WINDOW_SIZE+1
#define NBASIS     8             // GRID_SIZE + SPLINE_ORDER
#define COEFLEN    56            // INDIM * NBASIS
#define KANP       70            // COEFLEN + 2*INDIM
#define HH         192
#define WW         192
#define HWSZ       (HH*WW)       // 36864
#define BSZ        2
#define KDIM       279           // NCH * 9
#define KSTEPS     70            // ceil(279/4), covers k = 0..279 (padded)
#define NTILES     5             // ceil(70/16)
#define MTILES     ((BSZ*HWSZ)/16)   // 4608
#define BPACK_ELEMS (NTILES*KSTEPS*2*32)  // 22400

typedef float v2f __attribute__((ext_vector_type(2)));
typedef float v8f __attribute__((ext_vector_type(8)));

// Only meaningful to check amdgcn builtins during the DEVICE pass; the host
// pass (x86 clang) reports false for all amdgcn builtins.
#if defined(__HIP_DEVICE_COMPILE__)
#if !__has_builtin(__builtin_amdgcn_wmma_f32_16x16x4_f32)
#error "__builtin_amdgcn_wmma_f32_16x16x4_f32 not available in device compiler"
#endif
#endif

// ---------------------------------------------------------------------------
// Kernel 0: repack gen_w [70, 31, 3, 3] into WMMA B-fragment order:
//   Bpack[ntile][step][j][lane] = gen_w[n*279 + k]   (0 outside K<279, N<70)
//   with k = step*4 + j + 2*(lane>>4), n = ntile*16 + (lane&15)
// ---------------------------------------------------------------------------
__global__ void pack_weights_kernel(const float* __restrict__ gen_w,
                                    float* __restrict__ Bpack) {
    int t = blockIdx.x * blockDim.x + threadIdx.x;
    if (t >= BPACK_ELEMS) return;
    int lane  = t & 31;
    int j     = (t >> 5) & 1;
    int step  = (t >> 6) % KSTEPS;
    int ntile = t / (KSTEPS * 64);
    int k = step * 4 + j + 2 * (lane >> 4);
    int n = ntile * 16 + (lane & 15);
    float v = 0.0f;
    if (k < KDIM && n < KANP) v = gen_w[n * KDIM + k];
    Bpack[t] = v;
}

// ---------------------------------------------------------------------------
// Kernel 1: ctx[b,c,h,w] = mean over d of windowed_x[b,c,d,h,w]
// ---------------------------------------------------------------------------
__global__ void ctx_mean_kernel(const float* __restrict__ xin,
                                float* __restrict__ ctx) {
    int t = blockIdx.x * blockDim.x + threadIdx.x;
    if (t >= BSZ * NCH * HWSZ) return;
    int bc = t / HWSZ;
    int hw = t - bc * HWSZ;
    const float* p = xin + (size_t)bc * INDIM * HWSZ + hw;
    float s = 0.0f;
#pragma unroll
    for (int d = 0; d < INDIM; ++d) s += p[(size_t)d * HWSZ];
    ctx[t] = s * (1.0f / 7.0f);
}

// ---------------------------------------------------------------------------
// Kernel 2: implicit-GEMM 3x3 SAME conv via V_WMMA_F32_16X16X4_F32.
//   M = pixel (16/tile, tiles never straddle a row since 192%16==0)
//   K = c*9 + ky*3 + kx (padded to 280; channel 31 in LDS is zeros)
//   N = output KAN param (padded to 80; writes masked to n<70)
// Block: 160 threads = 5 waves; wave w owns N-tile w. LDS holds the 31x3x18
// ctx halo for the tile, shared by all 5 waves.
// ---------------------------------------------------------------------------
__global__ void __launch_bounds__(160)
conv_wmma_kernel(const float* __restrict__ ctx,
                 const float* __restrict__ Bpack,
                 const float* __restrict__ gen_b,
                 float* __restrict__ wts) {
    __shared__ float sctx[32][3][18];

    int mt   = blockIdx.x;
    int pix0 = mt * 16;
    int b    = pix0 / HWSZ;
    int rem  = pix0 - b * HWSZ;
    int hh   = rem / WW;
    int w0   = rem - hh * WW;

    // cooperative halo fill (zero padding at borders, zero channel 31)
    for (int e = threadIdx.x; e < 32 * 3 * 18; e += 160) {
        int c   = e / 54;
        int r   = (e / 18) % 3;
        int col = e - (e / 18) * 18;
        int gh  = hh + r - 1;
        int gw  = w0 + col - 1;
        float v = 0.0f;
        if (c < NCH && gh >= 0 && gh < HH && gw >= 0 && gw < WW)
            v = ctx[(size_t)(b * NCH + c) * HWSZ + gh * WW + gw];
        sctx[c][r][col] = v;
    }
    __syncthreads();

    int wid  = threadIdx.x >> 5;   // N-tile id 0..4
    int lane = threadIdx.x & 31;
    int hi   = lane >> 4;          // K half (and M half for D)
    int m    = lane & 15;

    v8f acc = {};
    const float* bp = Bpack + (size_t)wid * KSTEPS * 64;

    for (int step = 0; step < KSTEPS; ++step) {
        v2f afrag, bfrag;
#pragma unroll
        for (int j = 0; j < 2; ++j) {
            // A layout: VGPR j holds K = 4*step + j + 2*hi for row M = m
            int k   = step * 4 + j + 2 * hi;
            int c   = k / 9;
            int tap = k - 9 * c;
            int ky  = tap / 3;
            int kx  = tap - 3 * ky;
            afrag[j] = sctx[c][ky][m + kx];
            bfrag[j] = bp[step * 64 + j * 32 + lane];
        }
        acc = __builtin_amdgcn_wmma_f32_16x16x4_f32(
                  false, afrag, false, bfrag, (short)0, acc, false, false);
    }

    // D layout: VGPR r -> M = r + 8*hi, N = lane&15
    int n = wid * 16 + m;
    if (n < KANP) {
        float bias = gen_b[n];
#pragma unroll
        for (int r = 0; r < 8; ++r) {
            int M = r + 8 * hi;
            wts[(size_t)(pix0 + M) * KANP + n] = acc[r] + bias;
        }
    }
}

// ---------------------------------------------------------------------------
// Kernel 3: fused B-spline basis + spline/resid einsums.
// Block = 256 threads covering 8 pixels x 32 c-slots (c<31 active).
// Per-pixel 70 KAN params staged in LDS and shared across channels.
// ---------------------------------------------------------------------------
__global__ void __launch_bounds__(256)
kan_kernel(const float* __restrict__ xin,
           const float* __restrict__ wts,
           float* __restrict__ out) {
    __shared__ float swts[8][KANP];

    int blk  = blockIdx.x;
    int b    = blk / (HWSZ / 8);
    int hw0  = (blk - b * (HWSZ / 8)) * 8;
    int pix0 = b * HWSZ + hw0;

    for (int e = threadIdx.x; e < 8 * KANP; e += 256) {
        int pp = e / KANP;
        int kk = e - pp * KANP;
        swts[pp][kk] = wts[(size_t)(pix0 + pp) * KANP + kk];
    }
    __syncthreads();

    int c = threadIdx.x >> 3;
    int p = threadIdx.x & 7;
    if (c >= NCH) return;

    const float* s  = swts[p];
    const float* xp = xin + (size_t)(b * NCH + c) * INDIM * HWSZ + hw0 + p;

    // uniform knots: t_i = -1 + 0.4*(i-3), i = 0..11
    float t[12];
#pragma unroll
    for (int i = 0; i < 12; ++i) t[i] = -1.0f + 0.4f * (float)(i - 3);

    float accum = 0.0f;
#pragma unroll
    for (int d = 0; d < INDIM; ++d) {
        float x = xp[(size_t)d * HWSZ];

        // Cox-de Boor order-3 on uniform grid (exactly mirrors reference)
        float bb[11];
#pragma unroll
        for (int i = 0; i < 11; ++i)
            bb[i] = (x >= t[i] && x < t[i + 1]) ? 1.0f : 0.0f;
#pragma unroll
        for (int pp = 1; pp <= 3; ++pp) {
            float inv = 1.0f / (0.4f * (float)pp);
#pragma unroll
            for (int i = 0; i + pp < 11; ++i)
                bb[i] = (x - t[i]) * inv * bb[i] +
                        (t[i + pp + 1] - x) * inv * bb[i + 1];
        }

        float sd = 0.0f;
#pragma unroll
        for (int g = 0; g < NBASIS; ++g) sd += bb[g] * s[d * NBASIS + g];

        float sig = 1.0f / (1.0f + __expf(-x));          // silu = x*sigmoid
        accum += s[COEFLEN + d] * sd + s[COEFLEN + INDIM + d] * (x * sig);
    }

    out[(size_t)(b * NCH + c) * HWSZ + hw0 + p] = accum;
}

// ---------------------------------------------------------------------------
extern "C" void kernel_launch(void* const* d_in, const int* in_sizes, int n_in,
                              void* d_out, int out_size, void* d_ws, size_t ws_size,
                              hipStream_t stream) {
    (void)in_sizes; (void)n_in; (void)out_size; (void)ws_size;
    const float* xin   = (const float*)d_in[0];   // windowed_x [2,31,7,192,192]
    const float* gen_w = (const float*)d_in[1];   // [70,31,3,3]
    const float* gen_b = (const float*)d_in[2];   // [70]
    float* out = (float*)d_out;                   // [2,31,192,192]

    float* ws    = (float*)d_ws;
    float* Bpack = ws;                                   // 22400 floats
    float* ctx   = ws + 32768;                           // 2,285,568 floats
    float* wts   = ctx + (size_t)BSZ * NCH * HWSZ;       // 5,160,960 floats

    pack_weights_kernel<<<(BPACK_ELEMS + 255) / 256, 256, 0, stream>>>(gen_w, Bpack);
    ctx_mean_kernel<<<(BSZ * NCH * HWSZ + 255) / 256, 256, 0, stream>>>(xin, ctx);
    conv_wmma_kernel<<<MTILES, 160, 0, stream>>>(ctx, Bpack, gen_b, wts);
    kan_kernel<<<BSZ * (HWSZ / 8), 256, 0, stream>>>(xin, wts, out);
}